// MeanSquaredError3_35759897707069
// MI455X (gfx1250) — compile-verified
//
#include <hip/hip_runtime.h>
#include <hip/hip_bf16.h>
#include <math.h>

#define COLN  14
#define NJN   14
#define NCH   42          // 3 * NJ channels
#define PLANE 196         // 14*14

typedef float v2f __attribute__((ext_vector_type(2)));
typedef float v8f __attribute__((ext_vector_type(8)));

// G[m][c] for the Gaussian blur matrix built from symmetric-padded identity.
// gy[m] = G_MAT.T[yi][m] = G[m][yi]
__device__ __forceinline__ float gcol(int m, int c) {
    // raw (un-normalized) gaussian weights exp(-0.5*k^2), k = -4..4
    const float wraw[9] = {0.00033546262f, 0.011108997f, 0.13533528f, 0.60653067f,
                           1.0f,
                           0.60653067f, 0.13533528f, 0.011108997f, 0.00033546262f};
    const float invZ = 1.0f / 2.5066207842f; // sum of wraw
    float s = 0.0f;
#pragma unroll
    for (int k = 0; k < 9; ++k) {
        int r = k + m;                       // padded-row index
        int o = (r < 4) ? (3 - r) : ((r < 18) ? (r - 4) : (31 - r)); // symmetric pad map
        s += (o == c) ? wraw[k] * invZ : 0.0f;
    }
    return s;
}

__global__ void hm_loss_main(const float* __restrict__ outp,
                             const float* __restrict__ tp,
                             const float* __restrict__ vp,
                             float* __restrict__ partial,   // gridDim.x * 4 floats
                             int B) {
    __shared__ float  Glds[PLANE];   // G[m][c] at m*14+c
    __shared__ float2 cmm[COLN];     // per-column (min, max) of G
    __shared__ float  red[8][4];

    const int tid = threadIdx.x;

    // ---- one-time per-block: build G and per-column min/max in LDS
    if (tid < PLANE) {
        Glds[tid] = gcol(tid / COLN, tid - (tid / COLN) * COLN);
    }
    __syncthreads();
    if (tid < COLN) {
        float mn = Glds[tid], mx = Glds[tid];
        for (int m = 1; m < COLN; ++m) {
            float g = Glds[m * COLN + tid];
            mn = fminf(mn, g);
            mx = fmaxf(mx, g);
        }
        cmm[tid] = make_float2(mn, mx);
    }
    __syncthreads();

    const int lane = tid & 31;
    const int wv   = tid >> 5;
    const int wavesPerBlock = blockDim.x >> 5;
    const int gw = blockIdx.x * wavesPerBlock + wv;
    const int W  = gridDim.x * wavesPerBlock;
    const int P  = B * NJN;

    const int half = (lane >= 16) ? 1 : 0;   // which M-half this lane covers in C/D layout
    const int n    = lane & 15;              // N (column) index

    float acc1 = 0.f, acc2 = 0.f, accC = 0.f, accN = 0.f;

    for (int p = gw; p < P; p += W) {
        // p is wave-uniform: make it provably uniform so address math / t / v
        // processing scalarizes (SALU + s_load).
        const int pu = __builtin_amdgcn_readfirstlane(p);
        const int b  = pu / NJN;
        const int j  = pu - b * NJN;
        const float* hp = outp + ((size_t)b * NCH + j) * PLANE;

        // ---- load heatmap in WMMA 16x16 f32 C/D layout: reg r -> (m = r+8*half, n)
        float hv[8];
#pragma unroll
        for (int r = 0; r < 8; ++r) {
            int m = r + 8 * half;
            bool valid = (m < COLN) && (n < COLN);
            hv[r] = valid ? hp[m * COLN + n] : -__builtin_inff();
        }

        // ---- uniform per-pair scalars (scalar loads)
        const size_t tvBase = ((size_t)b * NJN + j) * 2;
        const float tx = tp[tvBase + 0];
        const float ty = tp[tvBase + 1];
        const float v0 = vp[tvBase + 0];
        const float v1 = vp[tvBase + 1];
        const int xi = (int)(tx * (float)COLN);
        const int yi = (int)(ty * (float)COLN);
        const bool inb  = (xi >= 0) && (xi <= COLN - 1) && (yi >= 0) && (yi <= COLN - 1);
        const bool vis  = ((int)v0) == 1;
        const bool scat = vis && inb;
        const float ve0 = (vis && !inb) ? 0.f : v0;
        const float ve1 = (vis && !inb) ? 0.f : v1;
        const bool visEff = ((int)ve0) == 1;
        const int yic = yi < 0 ? 0 : (yi > COLN - 1 ? COLN - 1 : yi);
        const int xic = xi < 0 ? 0 : (xi > COLN - 1 ? COLN - 1 : xi);

        // ---- Gaussian vector entries from LDS table
        const float gy_l = (n < COLN) ? Glds[n * COLN + yic] : 0.f;
        const float gx_l = (n < COLN) ? Glds[n * COLN + xic] : 0.f;

        // ---- normalization bounds from per-column tables (all entries >= 0)
        const float2 my = cmm[yic];
        const float2 mx2 = cmm[xic];
        const float mn  = my.x * mx2.x;
        const float mxv = my.y * mx2.y;
        const float inv = __builtin_amdgcn_rcpf(mxv - mn);

        // ---- rank-1 outer product tt = gy * gx^T via V_WMMA_F32_16X16X4_F32
        // A (16x4): column K=0 = gy, rest 0.  B (4x16): row K=0 = gx, rest 0.
        v2f Am, Bm;
        Am.x = half ? 0.f : gy_l;  Am.y = 0.f;
        Bm.x = half ? 0.f : gx_l;  Bm.y = 0.f;
        v8f Cz = {};
        v8f D = __builtin_amdgcn_wmma_f32_16x16x4_f32(
            /*neg_a=*/false, Am, /*neg_b=*/false, Bm,
            /*c_mod=*/(short)0, Cz, /*reuse_a=*/false, /*reuse_b=*/false);

        // ---- argmax over heatmap (first-max -> smallest flat index wins)
        float bestv = hv[0];
        int   besti = (8 * half) * COLN + n;
#pragma unroll
        for (int r = 1; r < 8; ++r) {
            int fi = (r + 8 * half) * COLN + n;
            if (hv[r] > bestv) { bestv = hv[r]; besti = fi; }
        }
#pragma unroll
        for (int mk = 1; mk < 32; mk <<= 1) {
            float ov = __shfl_xor(bestv, mk, 32);
            int   oi = __shfl_xor(besti, mk, 32);
            if (ov > bestv || (ov == bestv && oi < besti)) { bestv = ov; besti = oi; }
        }
        const int bi = __builtin_amdgcn_readfirstlane(besti); // uniform argmax index

        // ---- heatmap loss: sum (h - tt_norm)^2, row y==0 gated by visibility
        const float rowMask = visEff ? 1.f : 0.f;
        float s1 = 0.f;
#pragma unroll
        for (int r = 0; r < 8; ++r) {
            int m = r + 8 * half;
            bool valid = (m < COLN) && (n < COLN);
            float tte = scat ? (D[r] - mn) * inv : 0.f;
            float d   = valid ? (hv[r] - tte) : 0.f;
            if (m == 0) d *= rowMask;
            s1 += d * d;
        }
#pragma unroll
        for (int mk = 1; mk < 32; mk <<= 1) s1 += __shfl_xor(s1, mk, 32);
        // s1 now identical in all lanes

        // ---- coordinate loss: scalar gathers of ox/oy at the argmax index
        const float* oxp = outp + ((size_t)b * NCH + NJN + j) * PLANE;
        const float* oyp = outp + ((size_t)b * NCH + 2 * NJN + j) * PLANE;
        const float ox = oxp[bi];
        const float oy = oyp[bi];
        const int yc = bi / COLN;
        const int xc = bi - yc * COLN;
        const float scale = 1.0f / (float)COLN;
        const float px = (ox + (float)xc) * scale;
        const float py = (oy + (float)yc) * scale;
        const float dx = (px - tx) * ve0;
        const float dy = (py - ty) * ve1;

        // all quantities uniform across the wave: accumulate in every lane,
        // only lane 0's copy is written out below.
        acc1 += s1;
        acc2 += dx * dx + dy * dy;
        accC += visEff ? 1.f : 0.f;
        accN += (ve0 + ve1) * 0.5f;
    }

    // ---- block reduction of the 4 partial sums
    if (lane == 0 && wv < 8) {
        red[wv][0] = acc1; red[wv][1] = acc2; red[wv][2] = accC; red[wv][3] = accN;
    }
    __syncthreads();
    if (tid == 0) {
        float a = 0.f, bb = 0.f, c = 0.f, d = 0.f;
        for (int i = 0; i < wavesPerBlock; ++i) {
            a += red[i][0]; bb += red[i][1]; c += red[i][2]; d += red[i][3];
        }
        partial[(size_t)blockIdx.x * 4 + 0] = a;
        partial[(size_t)blockIdx.x * 4 + 1] = bb;
        partial[(size_t)blockIdx.x * 4 + 2] = c;
        partial[(size_t)blockIdx.x * 4 + 3] = d;
    }
}

__global__ void hm_loss_final(const float* __restrict__ partial, int nblocks,
                              float* __restrict__ result) {
    __shared__ float red[256][4];
    float a = 0.f, b = 0.f, c = 0.f, d = 0.f;
    for (int i = threadIdx.x; i < nblocks; i += blockDim.x) {
        a += partial[(size_t)i * 4 + 0];
        b += partial[(size_t)i * 4 + 1];
        c += partial[(size_t)i * 4 + 2];
        d += partial[(size_t)i * 4 + 3];
    }
    red[threadIdx.x][0] = a; red[threadIdx.x][1] = b;
    red[threadIdx.x][2] = c; red[threadIdx.x][3] = d;
    __syncthreads();
    if (threadIdx.x == 0) {
        float S1 = 0.f, S2 = 0.f, C = 0.f, N = 0.f;
        for (int i = 0; i < (int)blockDim.x; ++i) {
            S1 += red[i][0]; S2 += red[i][1]; C += red[i][2]; N += red[i][3];
        }
        result[0] = S1 / C + S2 / N;
    }
}

extern "C" void kernel_launch(void* const* d_in, const int* in_sizes, int n_in,
                              void* d_out, int out_size, void* d_ws, size_t ws_size,
                              hipStream_t stream) {
    const float* outp = (const float*)d_in[0];
    const float* tp   = (const float*)d_in[1];
    const float* vp   = (const float*)d_in[2];
    const int B = in_sizes[0] / (NCH * PLANE);

    float* partial = (float*)d_ws;           // needs blocks*4*4 = 16 KB
    const int blocks = 1024;                 // 8192 waves; ~14 pairs per wave

    hm_loss_main<<<blocks, 256, 0, stream>>>(outp, tp, vp, partial, B);
    hm_loss_final<<<1, 256, 0, stream>>>(partial, blocks, (float*)d_out);
}